// BiRNN_69320772157732
// MI455X (gfx1250) — compile-verified
//
#include <hip/hip_runtime.h>
#include <hip/hip_bf16.h>

// Bidirectional LSTM for MI455X (gfx1250, wave32).
//   Phase 1: xg = x @ W_ih^T + (b_ih + b_hh), both directions (backward uses
//            length-flipped time index folded into the load), full-GPU WMMA f32 GEMM.
//   Phase 2: persistent scan, 4 workgroups (2 dirs x 2 batch-halves, batches are
//            independent LSTMs). Each wave owns ONE 16x16 gate tile; its
//            W_hh^T slice lives in VGPRs (loaded once), h in LDS as float2
//            K-pairs (one ds_load_b64 per WMMA), c in registers.
// Workspace: xgF (128 MiB) + xgB (128 MiB) in d_ws.

typedef __attribute__((ext_vector_type(2))) float v2f;
typedef __attribute__((ext_vector_type(8))) float v8f;

#define T_LEN 2048
#define I_DIM 256
#define H_DIM 128
#define G_DIM 512            // 4*H
#define B_DIM 32
#define SCAN_B 16            // batch rows per scan workgroup (min WMMA M-tile)
#define HTP_STRIDE 17        // float2 slots per K-pair row (16 batches + pad)
#define G_STRIDE 516         // padded [16][516] gate staging in LDS

__device__ __forceinline__ float sigmoidf(float x) {
  return 1.0f / (1.0f + __expf(-x));
}

// ---------------------------------------------------------------------------
// Phase 1: xg[dir][b*T+t][g] = sum_i x[b][t_src][i] * W_ih[dir][g][i] + bias
// One 16-row x 64-col strip (4 WMMA tiles) per wave.
// ---------------------------------------------------------------------------
__global__ __launch_bounds__(256) void xg_gemm_kernel(
    const float* __restrict__ x, const int* __restrict__ lengths,
    const float* __restrict__ Wf, const float* __restrict__ bfi, const float* __restrict__ bfh,
    const float* __restrict__ Wb, const float* __restrict__ bbi, const float* __restrict__ bbh,
    float* __restrict__ xgF, float* __restrict__ xgB)
{
  const int lane = threadIdx.x & 31;
  const unsigned wid = blockIdx.x * 8u + (threadIdx.x >> 5);   // 65536 wave jobs
  const int n_strip = wid & 7;            // 8 strips of 64 cols
  const int m_tile  = (wid >> 3) & 4095;  // 4096 tiles of 16 rows
  const int dir     = wid >> 15;          // 0 = fwd, 1 = bwd

  const float* W  = dir ? Wb  : Wf;
  const float* bi = dir ? bbi : bfi;
  const float* bh = dir ? bbh : bfh;
  float* xg       = dir ? xgB : xgF;

  // A operand: 16 rows of x, with the padded-flip folded in for dir==1.
  const int b    = m_tile >> 7;                       // T/16 == 128 tiles per batch
  const int trow = ((m_tile & 127) << 4) + (lane & 15);
  int tsrc = trow;
  if (dir) { int L = lengths[b]; tsrc = (trow < L) ? (L - 1 - trow) : trow; }
  const int khalf = (lane >> 4) << 1;                 // lanes 16-31 hold K+2,K+3
  const float* arow = x + ((size_t)b * T_LEN + tsrc) * I_DIM + khalf;

  // B operand: W_ih rows (since B = W_ih^T), 4 col-tiles per wave.
  const int ncol = (n_strip << 6) + (lane & 15);
  const float* br0 = W + (size_t)(ncol +  0) * I_DIM + khalf;
  const float* br1 = W + (size_t)(ncol + 16) * I_DIM + khalf;
  const float* br2 = W + (size_t)(ncol + 32) * I_DIM + khalf;
  const float* br3 = W + (size_t)(ncol + 48) * I_DIM + khalf;

  v8f acc[4];
  acc[0] = (v8f){}; acc[1] = (v8f){}; acc[2] = (v8f){}; acc[3] = (v8f){};

#pragma unroll
  for (int kk = 0; kk < 64; ++kk) {
    const int kb = kk << 2;
    v2f a  = *(const v2f*)(arow + kb);
    v2f b0 = *(const v2f*)(br0 + kb);
    v2f b1 = *(const v2f*)(br1 + kb);
    v2f b2 = *(const v2f*)(br2 + kb);
    v2f b3 = *(const v2f*)(br3 + kb);
    acc[0] = __builtin_amdgcn_wmma_f32_16x16x4_f32(false, a, false, b0, (short)0, acc[0], false, false);
    acc[1] = __builtin_amdgcn_wmma_f32_16x16x4_f32(false, a, false, b1, (short)0, acc[1], false, false);
    acc[2] = __builtin_amdgcn_wmma_f32_16x16x4_f32(false, a, false, b2, (short)0, acc[2], false, false);
    acc[3] = __builtin_amdgcn_wmma_f32_16x16x4_f32(false, a, false, b3, (short)0, acc[3], false, false);
  }

  // C/D layout: reg r -> row m_tile*16 + r + 8*(lane>>4); col = base + (lane&15)
  const int mrowBase = (m_tile << 4) + ((lane >> 4) << 3);
#pragma unroll
  for (int tg = 0; tg < 4; ++tg) {
    const int col = (n_strip << 6) + (tg << 4) + (lane & 15);
    const float bias = bi[col] + bh[col];
#pragma unroll
    for (int r = 0; r < 8; ++r) {
      xg[(size_t)(mrowBase + r) * G_DIM + col] = acc[tg][r] + bias;
    }
  }
}

// ---------------------------------------------------------------------------
// Phase 2: persistent recurrent scan. 4 blocks: blockIdx = dir*2 + batch-half.
// Batches are independent, so each WG runs 16 of the 32 LSTM lanes.
// Wave w (0..31) owns gate-tile columns [16w, 16w+16); its W_hh^T slice is
// preloaded to VGPRs once and reused for all 2048 steps.
// ---------------------------------------------------------------------------
__global__ __launch_bounds__(1024) void lstm_scan_kernel(
    const float* __restrict__ xgF, const float* __restrict__ xgB,
    const float* __restrict__ Whh_f, const float* __restrict__ Whh_b,
    float* __restrict__ out)
{
  extern __shared__ float smem[];
  // htp: K-pair-major h^T: pair p (K = 2p, 2p+1) at htp[(p*HTP_STRIDE + m)*2 + {0,1}]
  float* htp = smem;                                   // 64 * 17 * 2 floats
  float* G   = smem + 64 * HTP_STRIDE * 2;             // [16][516] gate staging

  const int dir  = blockIdx.x >> 1;
  const int half = blockIdx.x & 1;
  const float* xg  = dir ? xgB : xgF;
  const float* Whh = dir ? Whh_b : Whh_f;

  const int tid  = threadIdx.x;
  const int lane = tid & 31;
  const int wave = tid >> 5;            // N-tile index (0..31)
  const int cb   = wave << 4;           // gate column base

  // zero initial hidden state
  for (int i = tid; i < 64 * HTP_STRIDE * 2; i += 1024) htp[i] = 0.0f;

  // Preload B = W_hh^T slice for this wave's tile: 32 k-steps x float2.
  const int khalf = (lane >> 4) << 1;
  const float* bp = Whh + (size_t)(cb + (lane & 15)) * H_DIM + khalf;
  v2f Breg[32];
#pragma unroll
  for (int kk = 0; kk < 32; ++kk) Breg[kk] = *(const v2f*)(bp + (kk << 2));

  // Elementwise ownership: thread -> batch row um (0..15), units jb, jb+64.
  const int um = tid & 15;
  const int jb = tid >> 4;              // 0..63
  float cst[2] = {0.0f, 0.0f};          // cell state lives in registers

  const int mA      = lane & 15;        // A-operand row (both half-waves)
  const int rowBase = (lane >> 4) << 3; // C/D row base per half-wave
  const int batch0  = half << 4;

  __syncthreads();

  for (int t = 0; t < T_LEN; ++t) {
    // Issue xg loads early; latency hides behind the WMMA K loop below.
    float xv[8];
#pragma unroll
    for (int r = 0; r < 8; ++r) {
      const int row = batch0 + rowBase + r;            // global batch index
      const float* p = xg + ((size_t)row * T_LEN + t) * G_DIM + cb + (lane & 15);
      xv[r] = *p;
      if (t + 1 < T_LEN) __builtin_prefetch(p + G_DIM, 0, 0);  // next step
    }

    // Recurrent GEMM: acc = h(t-1) @ W_hh^T for this wave's 16x16 tile.
    v8f acc = (v8f){};
#pragma unroll
    for (int kk = 0; kk < 32; ++kk) {
      const int pr = (kk << 1) + (lane >> 4);          // K-pair: {4kk..}+khalf
      v2f a = *(const v2f*)(htp + (pr * HTP_STRIDE + mA) * 2);
      acc = __builtin_amdgcn_wmma_f32_16x16x4_f32(false, a, false, Breg[kk], (short)0, acc, false, false);
    }

    // gates = recurrent + precomputed input projection (biases folded in xg)
#pragma unroll
    for (int r = 0; r < 8; ++r)
      G[(rowBase + r) * G_STRIDE + cb + (lane & 15)] = acc[r] + xv[r];
    __syncthreads();

    // Elementwise LSTM cell update; gate order along 4H is i, f, g, o.
#pragma unroll
    for (int q = 0; q < 2; ++q) {
      const int j = jb + (q << 6);
      const float gi = G[um * G_STRIDE + 0 * H_DIM + j];
      const float gf = G[um * G_STRIDE + 1 * H_DIM + j];
      const float gg = G[um * G_STRIDE + 2 * H_DIM + j];
      const float go = G[um * G_STRIDE + 3 * H_DIM + j];
      const float cn = sigmoidf(gf) * cst[q] + sigmoidf(gi) * tanhf(gg);
      cst[q] = cn;
      const float h = sigmoidf(go) * tanhf(cn);
      htp[((j >> 1) * HTP_STRIDE + um) * 2 + (j & 1)] = h;   // K index = j
      out[((size_t)(batch0 + um) * T_LEN + t) * (2 * H_DIM) + dir * H_DIM + j] = h;
    }
    __syncthreads();
  }
}

// ---------------------------------------------------------------------------
extern "C" void kernel_launch(void* const* d_in, const int* in_sizes, int n_in,
                              void* d_out, int out_size, void* d_ws, size_t ws_size,
                              hipStream_t stream) {
  const float* x     = (const float*)d_in[0];
  const int*   lens  = (const int*)  d_in[1];
  const float* Wf_ih = (const float*)d_in[2];
  const float* Wf_hh = (const float*)d_in[3];
  const float* bf_ih = (const float*)d_in[4];
  const float* bf_hh = (const float*)d_in[5];
  const float* Wb_ih = (const float*)d_in[6];
  const float* Wb_hh = (const float*)d_in[7];
  const float* bb_ih = (const float*)d_in[8];
  const float* bb_hh = (const float*)d_in[9];
  float* out = (float*)d_out;

  float* xgF = (float*)d_ws;                                  // 128 MiB
  float* xgB = xgF + (size_t)B_DIM * T_LEN * G_DIM;           // 128 MiB

  // 65536 wave-tiles (2 dirs x 4096 row-tiles x 8 col-strips), 8 waves/block.
  xg_gemm_kernel<<<8192, 256, 0, stream>>>(x, lens, Wf_ih, bf_ih, bf_hh,
                                           Wb_ih, bb_ih, bb_hh, xgF, xgB);

  const size_t smem = (size_t)(64 * HTP_STRIDE * 2 + SCAN_B * G_STRIDE) * sizeof(float);
  lstm_scan_kernel<<<4, 1024, smem, stream>>>(xgF, xgB, Wf_hh, Wb_hh, out);
}